// PerlinPowerFractalNoise_79654463472358
// MI455X (gfx1250) — compile-verified
//
#include <hip/hip_runtime.h>
#include <hip/hip_bf16.h>
#include <stdint.h>

// ---------------------------------------------------------------------------
// Perlin power-fractal noise, CDNA5 (gfx1250) implementation.
//   B=8, H=W=1024, SCALE=100, 8 octaves, persistence 0.5, lacunarity 2.
//   Pass 1: per-image 512-entry perm LUT staged into LDS via the CDNA5 async
//           global->LDS DMA path; branchless Perlin; clamped value stored to
//           d_out channel-0 slot; per-image min/max reduced via LDS atomics
//           (values in [0,1] => fp32 bits monotone as unsigned).
//   Pass 2: per-image min-max normalize, broadcast to RGB.
// ---------------------------------------------------------------------------

#define PPT 16            // pixels per thread in pass 1
#define BLK 256           // threads per block (8 wave32s per WGP dispatch)

__device__ __forceinline__ float fade_f(float t) {
    return t * t * t * (t * (t * 6.0f - 15.0f) + 10.0f);
}

__device__ __forceinline__ float lerp_f(float t, float a, float b) {
    return a + t * (b - a);
}

__device__ __forceinline__ float grad_f(int h, float x, float y, float z) {
    h &= 15;
    float u = (h < 8) ? x : y;
    float v = (h < 4) ? y : ((h == 12 || h == 14) ? x : z);
    float r = ((h & 1) == 0) ? u : -u;
    r += ((h & 2) == 0) ? v : -v;
    return r;
}

__device__ __forceinline__ float perlin3(float x, float y, float z, const int* p) {
    float fx = floorf(x), fy = floorf(y), fz = floorf(z);
    int X = ((int)fx) % 255;
    int Y = ((int)fy) % 255;
    int Z = ((int)fz) % 255;
    x -= fx; y -= fy; z -= fz;
    float u = fade_f(x), v = fade_f(y), w = fade_f(z);
    int A  = p[X] + Y;
    int AA = p[A] + Z;
    int AB = p[A + 1] + Z;
    int Bq = p[X + 1] + Y;
    int BA = p[Bq] + Z;
    int BB = p[Bq + 1] + Z;
    float l1 = lerp_f(u, grad_f(p[AA],     x,        y,        z),
                         grad_f(p[BA],     x - 1.0f, y,        z));
    float l2 = lerp_f(u, grad_f(p[AB],     x,        y - 1.0f, z),
                         grad_f(p[BB],     x - 1.0f, y - 1.0f, z));
    float l3 = lerp_f(u, grad_f(p[AA + 1], x,        y,        z - 1.0f),
                         grad_f(p[BA + 1], x - 1.0f, y,        z - 1.0f));
    float l4 = lerp_f(u, grad_f(p[AB + 1], x,        y - 1.0f, z - 1.0f),
                         grad_f(p[BB + 1], x - 1.0f, y - 1.0f, z - 1.0f));
    return lerp_f(w, lerp_f(v, l1, l2), lerp_f(v, l3, l4));
}

__global__ void perlin_init_minmax(unsigned* mm, int B) {
    int i = threadIdx.x;
    if (i < B) {
        mm[i]     = 0xFFFFFFFFu;  // running min (unsigned-compare of fp32 bits)
        mm[B + i] = 0u;           // running max
    }
}

__global__ void perlin_pass1(const float* __restrict__ xc,
                             const float* __restrict__ yc,
                             const float* __restrict__ zc,
                             const int* __restrict__ perm,
                             float* __restrict__ out,
                             unsigned* __restrict__ mm,
                             int B, int HW, int blocksPerImage) {
    __shared__ int lut[512];
    __shared__ unsigned sMin, sMax;

    const int tid   = threadIdx.x;
    const int b     = blockIdx.x / blocksPerImage;
    const int chunk = blockIdx.x % blocksPerImage;

    if (tid == 0) { sMin = 0xFFFFFFFFu; sMax = 0u; }

    // ---- CDNA5 async DMA: perm LUT (512 x i32 = 2KB) global -> LDS ----
    {
        const int* pbase = perm + b * 512;
        unsigned lds0 = (unsigned)(unsigned long long)(uintptr_t)&lut[0];
        for (int i = tid; i < 512; i += BLK) {
            unsigned la = lds0 + (unsigned)(i * 4);
            unsigned vo = (unsigned)(i * 4);
            asm volatile("global_load_async_to_lds_b32 %0, %1, %2"
                         :: "v"(la), "v"(vo), "s"(pbase)
                         : "memory");
        }
        asm volatile("s_wait_asynccnt 0" ::: "memory");
    }
    __syncthreads();  // LUT visible to all 8 waves; sMin/sMax initialized

    const int base = chunk * (PPT * BLK);
    float lmin =  3.0e38f;
    float lmax = -3.0e38f;

    for (int k = 0; k < PPT; ++k) {
        const int pix = base + k * BLK + tid;
        const int gi  = b * HW + pix;
        const float x = xc[gi];
        const float y = yc[gi];
        const float z = zc[gi];

        float total = 0.0f;
        float amp   = 1.0f;
        float freq  = 1.0f;
        float maxv  = 0.0f;
        #pragma unroll
        for (int o = 0; o < 8; ++o) {
            float xs = (x * freq) / 100.0f;
            float ys = (y * freq) / 100.0f;
            float zs = (z * freq) / 100.0f;
            total += amp * perlin3(xs, ys, zs, lut);
            maxv  += amp;
            freq  *= 2.0f;
            amp   *= 0.5f;
        }
        float n  = total / maxv;            // EXPONENT == 1.0 -> identity
        float nc = fminf(fmaxf(n, 0.0f), 1.0f);

        out[(size_t)gi * 3] = nc;           // stash in channel-0 slot
        lmin = fminf(lmin, nc);
        lmax = fmaxf(lmax, nc);
    }

    // in-LDS reduction (ds_min_u32 / ds_max_u32), then one global atomic/block
    atomicMin(&sMin, __float_as_uint(lmin));
    atomicMax(&sMax, __float_as_uint(lmax));
    __syncthreads();
    if (tid == 0) {
        atomicMin(&mm[b],     sMin);
        atomicMax(&mm[b + B], sMax);
    }
}

__global__ void perlin_pass2(float* __restrict__ out,
                             const unsigned* __restrict__ mm,
                             int B, int HW) {
    const int idx = blockIdx.x * blockDim.x + threadIdx.x;
    if (idx >= B * HW) return;
    const int b = idx / HW;
    const float mn = __uint_as_float(mm[b]);
    const float mx = __uint_as_float(mm[b + B]);
    const size_t o = (size_t)idx * 3;
    const float val  = out[o];
    const float norm = (val - mn) / (mx - mn);
    out[o]     = norm;
    out[o + 1] = norm;
    out[o + 2] = norm;
}

extern "C" void kernel_launch(void* const* d_in, const int* in_sizes, int n_in,
                              void* d_out, int out_size, void* d_ws, size_t ws_size,
                              hipStream_t stream) {
    const float* xc   = (const float*)d_in[0];
    const float* yc   = (const float*)d_in[1];
    const float* zc   = (const float*)d_in[2];
    const int*   perm = (const int*)d_in[3];
    float*       out  = (float*)d_out;
    unsigned*    mm   = (unsigned*)d_ws;   // needs 2*B*4 = 64 bytes of scratch

    const int B = 8, H = 1024, W = 1024;
    const int HW = H * W;
    const int blocksPerImage = HW / (PPT * BLK);   // 1048576 / 4096 = 256

    perlin_init_minmax<<<1, 32, 0, stream>>>(mm, B);
    perlin_pass1<<<B * blocksPerImage, BLK, 0, stream>>>(
        xc, yc, zc, perm, out, mm, B, HW, blocksPerImage);
    perlin_pass2<<<(B * HW + BLK - 1) / BLK, BLK, 0, stream>>>(out, mm, B, HW);
}